// GeNIEGQAAttention_69157563400748
// MI455X (gfx1250) — compile-verified
//
#include <hip/hip_runtime.h>
#include <hip/hip_bf16.h>
#include <cstdint>

// ---------------------------------------------------------------------------
// GeNIE GQA attention, fp32, CDNA5 (gfx1250), wave32 WMMA f32 16x16x4.
// B=2 P=128 C=8 T=1024 E=1024 H=16 KVH=4 G=4 D=64 EKV=256
// ---------------------------------------------------------------------------

#define BB   2
#define PP   128
#define CC   8
#define TT   1024          // P*C
#define EE   1024
#define HH   16
#define KVH  4
#define DD   64
#define EKV  256

typedef float v2f __attribute__((ext_vector_type(2)));
typedef float v8f __attribute__((ext_vector_type(8)));

__device__ __forceinline__ v8f wmma_f32(v2f a, v2f b, v8f c) {
  return __builtin_amdgcn_wmma_f32_16x16x4_f32(false, a, false, b, (short)0, c,
                                               false, false);
}

// ---------------------------------------------------------------------------
// Fast GEMM: Y[M,N] = A[M,K] @ Bw[K,N].  One wave per 16x64 tile (4 accums,
// A-fragment reused 4x).  M%16==0, N%64==0, K%8==0.  All loop-variant
// addresses are marching pointers (no per-iteration 64-bit muls).
// ---------------------------------------------------------------------------
__global__ void __launch_bounds__(32)
wgemm_f32_n64(const float* __restrict__ A, const float* __restrict__ Bw,
              float* __restrict__ Y, int M, int N, int K) {
  const int lane = threadIdx.x & 31;
  const int lm = lane & 15;
  const int half = lane >> 4;
  const int tn = blockIdx.x * 64;
  const int tm = blockIdx.y * 16;

  const float* aptr = A + (size_t)(tm + lm) * K + half * 2;
  const float* b0p  = Bw + (size_t)(half * 2) * N + tn + lm;
  const float* b1p  = b0p + N;
  const size_t bstep = (size_t)4 * N;

  v8f acc0 = {}, acc1 = {}, acc2 = {}, acc3 = {};
#pragma unroll 2
  for (int k = 0; k < K; k += 4) {
    const v2f a = *(const v2f*)aptr;                       // global_load_b64
    const v2f bb0 = {b0p[0],  b1p[0]};
    const v2f bb1 = {b0p[16], b1p[16]};
    const v2f bb2 = {b0p[32], b1p[32]};
    const v2f bb3 = {b0p[48], b1p[48]};
    acc0 = wmma_f32(a, bb0, acc0);
    acc1 = wmma_f32(a, bb1, acc1);
    acc2 = wmma_f32(a, bb2, acc2);
    acc3 = wmma_f32(a, bb3, acc3);
    aptr += 4;
    b0p += bstep;
    b1p += bstep;
  }

  float* yr = Y + (size_t)(tm + 8 * half) * N + tn + lm;
#pragma unroll
  for (int v = 0; v < 8; ++v) {
    yr[0]  = acc0[v];
    yr[16] = acc1[v];
    yr[32] = acc2[v];
    yr[48] = acc3[v];
    yr += N;
  }
}

// ---------------------------------------------------------------------------
// Ragged GEMM (bounds-checked M) — used only for the tiny ck projection (M=8).
// ---------------------------------------------------------------------------
__global__ void __launch_bounds__(32)
wgemm_f32_ragged(const float* __restrict__ A, const float* __restrict__ Bw,
                 float* __restrict__ Y, int M, int N, int K) {
  const int lane = threadIdx.x & 31;
  const int lm = lane & 15;
  const int half = lane >> 4;
  const int tn = blockIdx.x * 16;
  const int tm = blockIdx.y * 16;
  const int row = tm + lm;
  const bool rok = row < M;

  v8f acc = {};
  for (int k = 0; k < K; k += 4) {
    const int ka = k + half * 2;
    v2f a, b;
    a.x = rok ? A[(size_t)row * K + ka] : 0.0f;
    a.y = rok ? A[(size_t)row * K + ka + 1] : 0.0f;
    b.x = Bw[(size_t)ka * N + tn + lm];
    b.y = Bw[(size_t)(ka + 1) * N + tn + lm];
    acc = wmma_f32(a, b, acc);
  }
#pragma unroll
  for (int v = 0; v < 8; ++v) {
    const int r = tm + v + 8 * half;
    if (r < M) Y[(size_t)r * N + tn + lm] = acc[v];
  }
}

// ---------------------------------------------------------------------------
// Time scores: TA[bh, t, p] = sum_d (q[b,t,h*D+d] + tb[d]) * tk[p, kvh*D+d]
// grid = (P/64, T/16, B*H).  A fragments (q+bias) precomputed and reused
// across the 4 column sub-tiles; K = D = 64 fully unrolled (imm offsets).
// ---------------------------------------------------------------------------
__global__ void __launch_bounds__(32)
time_scores(const float* __restrict__ q, const float* __restrict__ tk,
            const float* __restrict__ bias, float* __restrict__ TA) {
  const int bh = blockIdx.z;
  const int b = bh >> 4, h = bh & 15, kvh = h >> 2;
  const int t0 = blockIdx.y * 16;
  const int p0 = blockIdx.x * 64;
  const int lane = threadIdx.x & 31;
  const int lm = lane & 15;
  const int half = lane >> 4;

  const float* qrow = q + ((size_t)(b * TT + t0 + lm)) * EE + h * DD + half * 2;
  const float* tb   = bias + EKV + kvh * DD + half * 2;

  v2f afrag[16];
#pragma unroll
  for (int k = 0; k < 16; ++k) {
    const v2f qv = *(const v2f*)(qrow + k * 4);
    const v2f bv = *(const v2f*)(tb + k * 4);
    afrag[k].x = qv.x + bv.x;
    afrag[k].y = qv.y + bv.y;
  }

  v8f acc[4] = {{}, {}, {}, {}};
#pragma unroll
  for (int j = 0; j < 4; ++j) {
    const float* tkc =
        tk + (size_t)(p0 + j * 16 + lm) * EKV + kvh * DD + half * 2;
#pragma unroll
    for (int k = 0; k < 16; ++k) {
      const v2f bfrag = *(const v2f*)(tkc + k * 4);        // contiguous pair
      acc[j] = wmma_f32(afrag[k], bfrag, acc[j]);
    }
  }
  float* orow = TA + ((size_t)bh * TT + t0 + 8 * half) * PP + p0 + lm;
#pragma unroll
  for (int v = 0; v < 8; ++v) {
#pragma unroll
    for (int j = 0; j < 4; ++j) orow[j * 16] = acc[j][v];
    orow += PP;
  }
}

// ---------------------------------------------------------------------------
// Transformer-XL relative shift (exact index math of _rel_shift):
//   TS[bh,t,p] = (k==0) ? 0 : TA_flat[bh, a*T + k-1]
//   with j = (t+1)*P + p, a = j/(T+1), k = j%(T+1)
// ---------------------------------------------------------------------------
__global__ void ts_shift(const float* __restrict__ TA, float* __restrict__ TS,
                         int total) {
  const int i = blockIdx.x * blockDim.x + threadIdx.x;
  if (i >= total) return;
  const int p = i & (PP - 1);
  const int t = (i >> 7) & (TT - 1);
  const int bh = i >> 17;
  const int j = (t + 1) * PP + p;
  const int a = j / (TT + 1);
  const int k = j % (TT + 1);
  float val = 0.0f;
  if (k != 0) val = TA[(size_t)bh * (TT * PP) + a * TT + (k - 1)];
  TS[i] = val;
}

// ---------------------------------------------------------------------------
// Channel scores with the channel rel-shift fused:
//   CS[bh,t,c] = sum_d (q[b,t,h*D+d] + cb[d]) * ck[C-1-|t%C - c|, kvh*D+d]
// ---------------------------------------------------------------------------
__global__ void chan_scores(const float* __restrict__ q,
                            const float* __restrict__ ck,
                            const float* __restrict__ bias,
                            float* __restrict__ CS, int total) {
  const int i = blockIdx.x * blockDim.x + threadIdx.x;
  if (i >= total) return;
  const int c = i & (CC - 1);
  const int t = (i >> 3) & (TT - 1);
  const int bh = i >> 13;
  const int b = bh >> 4, h = bh & 15, kvh = h >> 2;
  const int tm = t & (CC - 1);
  int diff = tm - c; if (diff < 0) diff = -diff;
  const int src = (CC - 1) - diff;

  const float4* q4  = (const float4*)(q + ((size_t)(b * TT + t)) * EE + h * DD);
  const float4* cb4 = (const float4*)(bias + 2 * EKV + kvh * DD);
  const float4* ck4 = (const float4*)(ck + (size_t)src * EKV + kvh * DD);
  float s = 0.0f;
#pragma unroll
  for (int d = 0; d < DD / 4; ++d) {
    const float4 a = q4[d], bb = cb4[d], cc = ck4[d];
    s += (a.x + bb.x) * cc.x + (a.y + bb.y) * cc.y +
         (a.z + bb.z) * cc.z + (a.w + bb.w) * cc.w;
  }
  CS[i] = s;
}

// ---------------------------------------------------------------------------
// Fused attention per (b,h, 16-row tile): band-masked WMMA QK^T (A fragments
// precomputed once), score assembly (TS + CS + masked global)/sqrt(D),
// causal mask, 32-lane softmax, causal-skipped WMMA PV from LDS with
// 4-way A reuse.  LDS: 16 x 1024 fp32 = 64 KB.
// ---------------------------------------------------------------------------
__global__ void __launch_bounds__(32)
attention(const float* __restrict__ q, const float* __restrict__ kv,
          const float* __restrict__ bias, const float* __restrict__ TS,
          const float* __restrict__ CS, float* __restrict__ ctx) {
  __shared__ float S[16 * TT];   // exactly 64 KB

  const int wg = blockIdx.x;          // bh*64 + tile
  const int tile = wg & 63;
  const int bh = wg >> 6;
  const int b = bh >> 4, h = bh & 15, kvh = h >> 2;
  const int t0 = tile * 16;
  const int lane = threadIdx.x & 31;
  const int lm = lane & 15;
  const int half = lane >> 4;

  const float* gb    = bias + kvh * DD + half * 2;
  const float* qrow  = q + ((size_t)(b * TT + t0 + lm)) * EE + h * DD + half * 2;
  const float* kvb   = kv + (size_t)b * TT * (2 * EKV);
  const float* tsrow = TS + (size_t)bh * TT * PP;
  const float* csrow = CS + (size_t)bh * TT * CC;

  const int tblo = t0 >> 3;
  const int tbhi = (t0 + 15) >> 3;

  // q + global-bias fragments, computed once, reused for all 64 s-tiles
  v2f afrag[16];
#pragma unroll
  for (int k = 0; k < 16; ++k) {
    const v2f qv = *(const v2f*)(qrow + k * 4);
    const v2f gv = *(const v2f*)(gb + k * 4);
    afrag[k].x = qv.x + gv.x;
    afrag[k].y = qv.y + gv.y;
  }

  // -------- scores --------
  const float* kbase = kvb + (size_t)lm * (2 * EKV) + kvh * DD + half * 2;
  for (int st = 0; st < 64; ++st) {
    const int sblo = 2 * st;
    const int sbhi = sblo + 1;
    const bool inBand = (sbhi >= tblo - 10) && (sblo <= tbhi + 10);
    const bool needG = (sblo <= tbhi) && (inBand || (tbhi == PP - 1));

    v8f acc = {};
    if (needG) {
      __builtin_prefetch(kbase + 16 * (2 * EKV), 0, 1);  // next s-tile row
#pragma unroll
      for (int k = 0; k < 16; ++k) {
        const v2f bfrag = *(const v2f*)(kbase + k * 4);
        acc = wmma_f32(afrag[k], bfrag, acc);
      }
    }
#pragma unroll
    for (int v = 0; v < 8; ++v) {
      const int r = v + 8 * half;
      const int t = t0 + r;
      const int s = st * 16 + lm;
      const int tb2 = t >> 3, sb = s >> 3;
      const bool causal = sb <= tb2;
      int bd = tb2 - sb; if (bd < 0) bd = -bd;
      const bool wmv = (bd <= 10) || (tb2 == PP - 1);
      float val = tsrow[(size_t)t * PP + sb] + csrow[(size_t)t * CC + (s & 7)] +
                  (wmv ? acc[v] : 0.0f);
      val *= 0.125f;                     // 1/sqrt(64)
      if (!causal) val = -3.40282347e38f;
      S[r * TT + s] = val;
    }
    kbase += 16 * (2 * EKV);
  }
  __syncthreads();

  // -------- softmax: 2 lanes per row (halves joined via shfl_xor 16) -------
  {
    float* rowp = S + lm * TT + half * (TT / 2);
    float mx = -3.40282347e38f;
    for (int s = 0; s < TT / 2; ++s) mx = fmaxf(mx, rowp[s]);
    mx = fmaxf(mx, __shfl_xor(mx, 16, 32));
    float sum = 0.0f;
    for (int s = 0; s < TT / 2; ++s) {
      const float e = __expf(rowp[s] - mx);
      rowp[s] = e;
      sum += e;
    }
    sum += __shfl_xor(sum, 16, 32);
    const float inv = 1.0f / sum;
    for (int s = 0; s < TT / 2; ++s) rowp[s] *= inv;
  }
  __syncthreads();

  // -------- PV: out[16,64] = W[16,T] @ V[T,64], causal-skipped.
  // st outer / 4 column accumulators inner: each ds_load_b64 A fragment
  // feeds 4 WMMAs; all addresses are marching base + immediate offsets.
  const int stMax = tbhi >> 1;           // last s-tile with any causal weight
  v8f pacc0 = {}, pacc1 = {}, pacc2 = {}, pacc3 = {};
  {
    const float* vbase =
        kvb + EKV + kvh * DD + (size_t)(half * 2) * (2 * EKV) + lm;
    const float* sbase = S + lm * TT + half * 2;
    for (int st = 0; st <= stMax; ++st) {
#pragma unroll
      for (int ks = 0; ks < 4; ++ks) {
        const v2f a = *(const v2f*)(sbase + ks * 4);       // ds_load_b64
        const float* vr = vbase + (size_t)(ks * 4) * (2 * EKV);
        const v2f b0 = {vr[0],  vr[2 * EKV]};
        const v2f b1 = {vr[16], vr[2 * EKV + 16]};
        const v2f b2 = {vr[32], vr[2 * EKV + 32]};
        const v2f b3 = {vr[48], vr[2 * EKV + 48]};
        pacc0 = wmma_f32(a, b0, pacc0);
        pacc1 = wmma_f32(a, b1, pacc1);
        pacc2 = wmma_f32(a, b2, pacc2);
        pacc3 = wmma_f32(a, b3, pacc3);
      }
      sbase += 16;
      vbase += (size_t)16 * (2 * EKV);
    }
  }
  {
    float* orow = ctx + ((size_t)(b * TT + t0 + 8 * half)) * EE + h * DD + lm;
#pragma unroll
    for (int v = 0; v < 8; ++v) {
      orow[0]  = pacc0[v];
      orow[16] = pacc1[v];
      orow[32] = pacc2[v];
      orow[48] = pacc3[v];
      orow += EE;
    }
  }
}

// ---------------------------------------------------------------------------
// Host-side orchestration
// ---------------------------------------------------------------------------
extern "C" void kernel_launch(void* const* d_in, const int* in_sizes, int n_in,
                              void* d_out, int out_size, void* d_ws,
                              size_t ws_size, hipStream_t stream) {
  const float* X     = (const float*)d_in[0];  // (B*T, E)
  const float* Pos   = (const float*)d_in[1];  // (P, E)
  const float* Chan  = (const float*)d_in[2];  // (C, E)
  const float* Wq    = (const float*)d_in[3];  // (E, E)
  const float* Wkv   = (const float*)d_in[4];  // (E, 2*EKV)
  const float* Wpos  = (const float*)d_in[5];  // (E, EKV)
  const float* Wchan = (const float*)d_in[6];  // (E, EKV)
  const float* Wproj = (const float*)d_in[7];  // (E, E)
  const float* bias  = (const float*)d_in[8];  // (3*EKV,)
  float* out = (float*)d_out;

  float* ws = (float*)d_ws;
  const size_t nQ  = (size_t)BB * TT * EE;          // 2,097,152
  const size_t nKV = (size_t)BB * TT * 2 * EKV;     // 1,048,576
  const size_t nTK = (size_t)PP * EKV;              //    32,768
  const size_t nCK = (size_t)CC * EKV;              //     2,048
  const size_t nTA = (size_t)BB * HH * TT * PP;     // 4,194,304
  const size_t nCS = (size_t)BB * HH * TT * CC;     //   262,144

  float* qbuf  = ws;               ws += nQ;
  float* kvbuf = ws;               ws += nKV;
  float* tkbuf = ws;               ws += nTK;
  float* ckbuf = ws;               ws += nCK;
  float* TA    = ws;               ws += nTA;
  float* TS    = ws;               ws += nTA;
  float* CSb   = ws;               ws += nCS;
  float* ctx   = ws;               ws += nQ;

  const int Mrows = BB * TT;  // 2048

  // Projections (all M,N tile-exact except ck)
  wgemm_f32_n64<<<dim3(EE / 64, Mrows / 16), 32, 0, stream>>>(X, Wq, qbuf, Mrows, EE, EE);
  wgemm_f32_n64<<<dim3((2 * EKV) / 64, Mrows / 16), 32, 0, stream>>>(X, Wkv, kvbuf, Mrows, 2 * EKV, EE);
  wgemm_f32_n64<<<dim3(EKV / 64, PP / 16), 32, 0, stream>>>(Pos, Wpos, tkbuf, PP, EKV, EE);
  wgemm_f32_ragged<<<dim3(EKV / 16, 1), 32, 0, stream>>>(Chan, Wchan, ckbuf, CC, EKV, EE);

  // Time scores + relative shift
  time_scores<<<dim3(PP / 64, TT / 16, BB * HH), 32, 0, stream>>>(qbuf, tkbuf, bias, TA);
  {
    const int total = (int)nTA;
    ts_shift<<<(total + 255) / 256, 256, 0, stream>>>(TA, TS, total);
  }

  // Channel scores (shift fused)
  {
    const int total = (int)nCS;
    chan_scores<<<(total + 255) / 256, 256, 0, stream>>>(qbuf, ckbuf, bias, CSb, total);
  }

  // Fused attention
  attention<<<dim3(BB * HH * (TT / 16)), 32, 0, stream>>>(qbuf, kvbuf, bias, TS, CSb, ctx);

  // Output projection
  wgemm_f32_n64<<<dim3(EE / 64, Mrows / 16), 32, 0, stream>>>(ctx, Wproj, out, Mrows, EE, EE);

  (void)in_sizes; (void)n_in; (void)out_size; (void)ws_size;
}